// GNN_22282290331959
// MI455X (gfx1250) — compile-verified
//
#include <hip/hip_runtime.h>
#include <math.h>

// ---------------------------------------------------------------------------
// GNN forward (TransformerConv x5 + TopK pool x4 + MLP) for MI455X (gfx1250).
// GEMMs use V_WMMA_F32_16X16X4_F32; each wave computes a 16x64 strip (4 WMMA
// accumulators) with a guard-free main K loop and a branch-free select tail.
// Segment softmax over edges uses uint-keyed float atomicMax + float atomicAdd.
// ---------------------------------------------------------------------------

#define HH   3
#define DD   128
#define HD   384          // H*D
#define NB   256          // batch B
#define NE   512          // edges per graph
#define BEC  (NB*NE)      // 131072 total edges
#define EPSC 1e-5f

typedef float v2f __attribute__((ext_vector_type(2)));
typedef float v8f __attribute__((ext_vector_type(8)));

// ---- monotone float<->uint key for atomic float max ------------------------
__device__ __forceinline__ unsigned fkey(float x) {
  unsigned u = __float_as_uint(x);
  return (u & 0x80000000u) ? ~u : (u | 0x80000000u);
}
__device__ __forceinline__ float funkey(unsigned k) {
  return (k & 0x80000000u) ? __uint_as_float(k & 0x7fffffffu)
                           : __uint_as_float(~k);
}

#define WMMA_F32(a, b, c) \
  __builtin_amdgcn_wmma_f32_16x16x4_f32(false, (a), false, (b), (short)0, (c), false, false)

// ---- WMMA f32 GEMM: Y[M,N] = act(X[M,K] @ W[K,N] + bias) -------------------
// One wave per 16(M) x 64(N) strip: 4 v8f accumulators, 4 WMMA per k-step.
// A frag: lane L<16 holds row L, K k0..k0+1; lane>=16 holds row L-16, K k0+2..3
// B frag: lane L holds col L%16; vgpr r = K row k0+r (+2 for high lanes)
// C frag: vgpr r, lane L<16 -> row r, col L%16; lane>=16 -> row r+8
// Main loop is guard-free; K tail uses clamped addresses + selects (no branches).
__global__ void gemm_wmma_f32(const float* __restrict__ X,
                              const float* __restrict__ W,
                              const float* __restrict__ bias,
                              float* __restrict__ Y,
                              int M, int N, int K, int act) {
  int ntn  = N >> 6;
  int tile = blockIdx.x;
  int tm   = (tile / ntn) << 4;
  int tn   = (tile % ntn) << 6;
  int lane = threadIdx.x & 31;
  int lo   = lane & 15;
  int hiK  = (lane >> 4) << 1;   // 0 or 2
  const float* Xr = X + (size_t)(tm + lo) * K;
  const float* Wc = W + tn + lo;
  v8f c0 = {}, c1 = {}, c2 = {}, c3 = {};
  int K4 = K & ~3;
  for (int k0 = 0; k0 < K4; k0 += 4) {
    int ka = k0 + hiK;
    v2f a;
    a.x = Xr[ka];
    a.y = Xr[ka + 1];
    const float* w0 = Wc + (size_t)ka * N;
    const float* w1 = Wc + (size_t)(ka + 1) * N;
    v2f b0, b1, b2, b3;
    b0.x = w0[0];  b0.y = w1[0];
    b1.x = w0[16]; b1.y = w1[16];
    b2.x = w0[32]; b2.y = w1[32];
    b3.x = w0[48]; b3.y = w1[48];
    c0 = WMMA_F32(a, b0, c0);
    c1 = WMMA_F32(a, b1, c1);
    c2 = WMMA_F32(a, b2, c2);
    c3 = WMMA_F32(a, b3, c3);
  }
  if (K4 < K) {                      // ragged K tail (K=30, K=11): branch-free
    int ka = K4 + hiK;
    float m0 = (ka     < K) ? 1.f : 0.f;
    float m1 = (ka + 1 < K) ? 1.f : 0.f;
    int  i0 = (ka     < K) ? ka     : 0;   // clamped (always valid) addresses
    int  i1 = (ka + 1 < K) ? ka + 1 : 0;
    v2f a;
    a.x = Xr[i0] * m0;
    a.y = Xr[i1] * m1;
    const float* w0 = Wc + (size_t)i0 * N;
    const float* w1 = Wc + (size_t)i1 * N;
    v2f b0, b1, b2, b3;
    b0.x = w0[0]  * m0; b0.y = w1[0]  * m1;
    b1.x = w0[16] * m0; b1.y = w1[16] * m1;
    b2.x = w0[32] * m0; b2.y = w1[32] * m1;
    b3.x = w0[48] * m0; b3.y = w1[48] * m1;
    c0 = WMMA_F32(a, b0, c0);
    c1 = WMMA_F32(a, b1, c1);
    c2 = WMMA_F32(a, b2, c2);
    c3 = WMMA_F32(a, b3, c3);
  }
  int rb = tm + ((lane >> 4) << 3);
  float g0 = bias ? bias[tn + lo]      : 0.f;
  float g1 = bias ? bias[tn + lo + 16] : 0.f;
  float g2 = bias ? bias[tn + lo + 32] : 0.f;
  float g3 = bias ? bias[tn + lo + 48] : 0.f;
#pragma unroll
  for (int r = 0; r < 8; ++r) {
    float* yr = Y + (size_t)(rb + r) * N + tn + lo;
    float v0 = c0[r] + g0, v1 = c1[r] + g1, v2 = c2[r] + g2, v3 = c3[r] + g3;
    if (act) {
      v0 = fmaxf(v0, 0.f); v1 = fmaxf(v1, 0.f);
      v2 = fmaxf(v2, 0.f); v3 = fmaxf(v3, 0.f);
    }
    yr[0] = v0; yr[16] = v1; yr[32] = v2; yr[48] = v3;
  }
}

// ---- edge list init --------------------------------------------------------
__global__ void init_edges_k(const int* __restrict__ es, const int* __restrict__ ed,
                             int* sl, int* dl, int* em) {
  int i = blockIdx.x * blockDim.x + threadIdx.x;
  if (i < BEC) { sl[i] = es[i]; dl[i] = ed[i]; em[i] = 1; }
}

__global__ void init_seg_k(unsigned* am, float* den, int cnt) {
  int i = blockIdx.x * blockDim.x + threadIdx.x;
  if (i < cnt) { am[i] = 0x007FFFFFu /* key(-inf) */; den[i] = 0.f; }
}

// ---- alpha = (q[dst] . (k[src]+e)) / sqrt(D); segment max over dst ---------
__global__ void alpha_k(const float* __restrict__ q, const float* __restrict__ kk,
                        const float* __restrict__ e, const int* __restrict__ sl,
                        const int* __restrict__ dl, const int* __restrict__ em,
                        float* __restrict__ alpha, unsigned* __restrict__ amax,
                        int n) {
  int w    = blockIdx.x * (blockDim.x >> 5) + (threadIdx.x >> 5);
  int lane = threadIdx.x & 31;
  if (w >= BEC * HH) return;
  int j = w / HH, h = w - j * HH;
  int b = j / NE;
  int s = sl[j] + b * n;
  int d = dl[j] + b * n;
  const float* qp = q  + (size_t)d * HD + h * DD;
  const float* kp = kk + (size_t)s * HD + h * DD;
  const float* ep = e  + (size_t)j * HD + h * DD;
  float p = 0.f;
  for (int t = lane; t < DD; t += 32) p += qp[t] * (kp[t] + ep[t]);
  for (int o = 16; o; o >>= 1) p += __shfl_down(p, o, 32);
  if (lane == 0) {
    int m = em[j];
    float a = p * 0.088388347648318447f;  // 1/sqrt(128)
    alpha[(size_t)j * HH + h] = m ? a : -__builtin_inff();
    if (m) atomicMax(&amax[(size_t)d * HH + h], fkey(a));
  }
}

// ---- pexp (in place over alpha) + segment-sum denom ------------------------
__global__ void pexp_k(float* __restrict__ alpha, const unsigned* __restrict__ amax,
                       float* __restrict__ den, const int* __restrict__ dl,
                       const int* __restrict__ em, int n) {
  int i = blockIdx.x * blockDim.x + threadIdx.x;
  if (i >= BEC * HH) return;
  int j = i / HH, h = i - j * HH;
  int b = j / NE;
  int d = dl[j] + b * n;
  float am = funkey(amax[(size_t)d * HH + h]);
  if (am == -__builtin_inff()) am = 0.f;   // where(isfinite, amax, 0)
  float p = em[j] ? expf(alpha[i] - am) : 0.f;
  alpha[i] = p;
  if (p != 0.f) atomicAdd(&den[(size_t)d * HH + h], p);
}

// ---- agg[dst] += (v[src]+e) * pexp ----------------------------------------
__global__ void agg_k(const float* __restrict__ v, const float* __restrict__ e,
                      const float* __restrict__ pexp, const int* __restrict__ sl,
                      const int* __restrict__ dl, float* __restrict__ agg, int n) {
  unsigned i = blockIdx.x * blockDim.x + threadIdx.x;
  if (i >= (unsigned)BEC * HD) return;
  int j = i / HD, cc = i - (unsigned)j * HD, h = cc >> 7;
  float p = pexp[(size_t)j * HH + h];
  if (p == 0.f) return;
  int b = j / NE;
  int s = sl[j] + b * n, d = dl[j] + b * n;
  atomicAdd(&agg[(size_t)d * HD + cc],
            (v[(size_t)s * HD + cc] + e[(size_t)j * HD + cc]) * p);
}

// ---- beta-gated combine: out = beta*skip + (1-beta)*(agg/denom) ------------
__global__ void gate_k(const float* __restrict__ agg, const float* __restrict__ skip,
                       const float* __restrict__ den, const float* __restrict__ Wb,
                       float* __restrict__ out, int M) {
  int w    = blockIdx.x * (blockDim.x >> 5) + (threadIdx.x >> 5);
  int lane = threadIdx.x & 31;
  if (w >= M) return;
  float an[12], sk[12], bsum = 0.f;
#pragma unroll
  for (int i = 0; i < 12; ++i) {
    int t = lane + i * 32;
    int h = t >> 7;
    float dv = fmaxf(den[(size_t)w * HH + h], 1e-16f);
    float a  = agg[(size_t)w * HD + t] / dv;
    float s  = skip[(size_t)w * HD + t];
    an[i] = a; sk[i] = s;
    bsum += a * Wb[t] + s * Wb[HD + t] + (a - s) * Wb[2 * HD + t];
  }
  for (int o = 16; o; o >>= 1) bsum += __shfl_xor(bsum, o, 32);
  float beta = 1.f / (1.f + expf(-bsum));
#pragma unroll
  for (int i = 0; i < 12; ++i) {
    int t = lane + i * 32;
    out[(size_t)w * HD + t] = beta * sk[i] + (1.f - beta) * an[i];
  }
}

// ---- batchnorm -------------------------------------------------------------
__global__ void bn_stats_k(const float* __restrict__ x, float* __restrict__ st, int M) {
  int f = threadIdx.x;                 // feature 0..127
  float s = 0.f, ss = 0.f;
  for (int m = blockIdx.x; m < M; m += gridDim.x) {
    float v = x[(size_t)m * DD + f];
    s += v; ss += v * v;
  }
  atomicAdd(&st[f], s);
  atomicAdd(&st[DD + f], ss);
}

__global__ void bn_apply_k(float* __restrict__ x, const float* __restrict__ st,
                           const float* __restrict__ g, const float* __restrict__ b,
                           int M) {
  unsigned i = blockIdx.x * blockDim.x + threadIdx.x;
  if (i >= (unsigned)M * DD) return;
  int f = i & (DD - 1);
  float mu  = st[f] / M;
  float var = st[DD + f] / M - mu * mu;
  x[i] = g[f] * (x[i] - mu) * rsqrtf(var + EPSC) + b[f];
}

// ---- top-k pooling ---------------------------------------------------------
__global__ void wnorm_k(const float* __restrict__ w, float* __restrict__ out) {
  __shared__ float sh[DD];
  int t = threadIdx.x;
  sh[t] = w[t] * w[t];
  __syncthreads();
  for (int o = 64; o; o >>= 1) { if (t < o) sh[t] += sh[t + o]; __syncthreads(); }
  if (t == 0) out[0] = sqrtf(sh[0]);
}

__global__ void score_k(const float* __restrict__ x, const float* __restrict__ w,
                        const float* __restrict__ wn, float* __restrict__ sc, int M) {
  int wi   = blockIdx.x * (blockDim.x >> 5) + (threadIdx.x >> 5);
  int lane = threadIdx.x & 31;
  if (wi >= M) return;
  float p = 0.f;
  for (int t = lane; t < DD; t += 32) p += x[(size_t)wi * DD + t] * w[t];
  for (int o = 16; o; o >>= 1) p += __shfl_down(p, o, 32);
  if (lane == 0) sc[wi] = tanhf(p / wn[0]);
}

__global__ void topk_k(const float* __restrict__ sc, int* __restrict__ idx,
                       float* __restrict__ vals, int* __restrict__ inv,
                       int n, int k) {
  __shared__ float sh[128];
  int b = blockIdx.x, t = threadIdx.x;
  if (t < n) { sh[t] = sc[b * n + t]; inv[b * n + t] = -1; }
  __syncthreads();
  if (t == 0) {
    for (int i = 0; i < k; ++i) {        // stable argmax: ties -> lower index
      int best = 0; float bv = sh[0];
      for (int j = 1; j < n; ++j) if (sh[j] > bv) { bv = sh[j]; best = j; }
      idx[b * k + i]  = best;
      vals[b * k + i] = bv;
      inv[b * n + best] = i;
      sh[best] = -__builtin_inff();
    }
  }
}

__global__ void pool_gather_k(const float* __restrict__ xin, const int* __restrict__ idx,
                              const float* __restrict__ vals, float* __restrict__ xout,
                              int n, int k) {
  int bi = blockIdx.x;                  // b*k + i
  int d  = threadIdx.x;                 // 0..127
  int b  = bi / k;
  int sn = idx[bi];
  xout[(size_t)bi * DD + d] = xin[(size_t)(b * n + sn) * DD + d] * vals[bi];
}

__global__ void remap_k(int* sl, int* dl, int* em, const int* __restrict__ inv, int n) {
  int j = blockIdx.x * blockDim.x + threadIdx.x;
  if (j >= BEC) return;
  int b  = j / NE;
  int ns = inv[b * n + sl[j]];
  int nd = inv[b * n + dl[j]];
  em[j] = em[j] && (ns >= 0) && (nd >= 0);
  sl[j] = ns > 0 ? ns : 0;
  dl[j] = nd > 0 ? nd : 0;
}

__global__ void reps_k(const float* __restrict__ x, float* __restrict__ h, int k) {
  int b = blockIdx.x, d = threadIdx.x;
  float mx = -__builtin_inff(), sm = 0.f;
  for (int i = 0; i < k; ++i) {
    float v = x[(size_t)(b * k + i) * DD + d];
    mx = fmaxf(mx, v);
    sm += v;
  }
  h[b * 3 * DD + d]            += mx;
  h[b * 3 * DD + DD + d]       += sm / k;
  h[b * 3 * DD + 2 * DD + d]   += sm;
}

// ---- final N=1 projection --------------------------------------------------
__global__ void rowdot_k(const float* __restrict__ x, const float* __restrict__ w,
                         const float* __restrict__ bias, float* __restrict__ out,
                         int M, int K) {
  int r    = blockIdx.x * (blockDim.x >> 5) + (threadIdx.x >> 5);
  int lane = threadIdx.x & 31;
  if (r >= M) return;
  float p = 0.f;
  for (int t = lane; t < K; t += 32) p += x[(size_t)r * K + t] * w[t];
  for (int o = 16; o; o >>= 1) p += __shfl_down(p, o, 32);
  if (lane == 0) out[r] = p + bias[0];
}

// ---------------------------------------------------------------------------
extern "C" void kernel_launch(void* const* d_in, const int* in_sizes, int n_in,
                              void* d_out, int out_size, void* d_ws, size_t ws_size,
                              hipStream_t stream) {
  (void)in_sizes; (void)n_in; (void)out_size; (void)ws_size;
  const float* x0    = (const float*)d_in[0];       // (256,128,30)
  const float* eattr = (const float*)d_in[1];       // (256,512,11)
  const int*   esrc  = (const int*)  d_in[2];
  const int*   edst  = (const int*)  d_in[3];
  auto P = [&](int i) { return (const float*)d_in[i]; };

  // jax pytree order (sorted dict keys):
  // 4:bn1.b 5:bn1.g 6..15:conv1{Wb,We,Wk,Wq,Ws,Wv,bk,bq,bs,bv}
  // 16+15i: layer i {bn.b,bn.g, conv{...x10}, pool_w, transf.W, transf.b}
  // 76..81: lin1.W,b lin2.W,b lin3.W,b   82,83: transf1.W,b
  struct Conv { const float *Wb,*We,*Wk,*Wq,*Ws,*Wv,*bk,*bq,*bs,*bv; };
  Conv cv[5];
  const float *bng[5], *bnb[5], *tW[5], *tb[5], *poolw[4];
  bnb[0]=P(4); bng[0]=P(5);
  cv[0]={P(6),P(7),P(8),P(9),P(10),P(11),P(12),P(13),P(14),P(15)};
  tW[0]=P(82); tb[0]=P(83);
  for (int i=0;i<4;++i) {
    int b0 = 16 + i*15;
    bnb[i+1]=P(b0+0); bng[i+1]=P(b0+1);
    cv[i+1]={P(b0+2),P(b0+3),P(b0+4),P(b0+5),P(b0+6),P(b0+7),
             P(b0+8),P(b0+9),P(b0+10),P(b0+11)};
    poolw[i]=P(b0+12); tW[i+1]=P(b0+13); tb[i+1]=P(b0+14);
  }
  const float *l1W=P(76), *l1b=P(77), *l2W=P(78), *l2b=P(79), *l3W=P(80), *l3b=P(81);

  // ---- workspace carve-out (floats; 256B aligned) ----
  size_t off = 0;
  auto alloc = [&](size_t elems) {
    void* p = (char*)d_ws + off;
    off += ((elems * 4 + 255) & ~(size_t)255);
    return p;
  };
  const int MMAX = NB * 128;                       // 32768
  float* q    = (float*)alloc((size_t)MMAX * HD);
  float* kbuf = (float*)alloc((size_t)MMAX * HD);
  float* vbuf = (float*)alloc((size_t)MMAX * HD);
  float* skip = (float*)alloc((size_t)MMAX * HD);
  float* agg  = (float*)alloc((size_t)MMAX * HD);
  float* ebuf = (float*)alloc((size_t)BEC  * HD);  // edge projections
  float* alpha= (float*)alloc((size_t)BEC  * HH);  // reused as pexp in-place
  unsigned* amax = (unsigned*)alloc((size_t)MMAX * HH);
  float* den  = (float*)alloc((size_t)MMAX * HH);
  float* xA   = (float*)alloc((size_t)MMAX * DD);
  float* xB   = (float*)alloc((size_t)MMAX * DD);
  int*   sl   = (int*)  alloc(BEC);
  int*   dl   = (int*)  alloc(BEC);
  int*   em   = (int*)  alloc(BEC);
  float* sc   = (float*)alloc((size_t)NB * 128);
  int*   idxk = (int*)  alloc((size_t)NB * 64);
  float* vals = (float*)alloc((size_t)NB * 64);
  int*   inv  = (int*)  alloc((size_t)NB * 128);
  float* st   = (float*)alloc(2 * DD);
  float* wn   = (float*)alloc(1);
  float* hrep = (float*)alloc((size_t)NB * 3 * DD);
  float* m1   = (float*)alloc((size_t)NB * 256);
  float* m2   = (float*)alloc((size_t)NB * DD);

  auto gemm = [&](const float* X, const float* W, const float* bias, float* Y,
                  int M, int N, int K, int act) {
    gemm_wmma_f32<<<(M >> 4) * (N >> 6), 32, 0, stream>>>(X, W, bias, Y, M, N, K, act);
  };

  init_edges_k<<<BEC / 256, 256, 0, stream>>>(esrc, edst, sl, dl, em);
  hipMemsetAsync(hrep, 0, (size_t)NB * 3 * DD * 4, stream);

  int n = 128;
  const float* xin = x0;
  int K = 30;
  for (int blk = 0; blk < 5; ++blk) {
    int M = NB * n;
    float* xnext = (xin == xA) ? xB : xA;

    gemm(xin, cv[blk].Wq, cv[blk].bq, q,    M, HD, K, 0);
    gemm(xin, cv[blk].Wk, cv[blk].bk, kbuf, M, HD, K, 0);
    gemm(xin, cv[blk].Wv, cv[blk].bv, vbuf, M, HD, K, 0);
    gemm(xin, cv[blk].Ws, cv[blk].bs, skip, M, HD, K, 0);
    gemm(eattr, cv[blk].We, nullptr, ebuf, BEC, HD, 11, 0);

    init_seg_k<<<(M * HH + 255) / 256, 256, 0, stream>>>(amax, den, M * HH);
    alpha_k<<<(BEC * HH) / 4, 128, 0, stream>>>(q, kbuf, ebuf, sl, dl, em,
                                                alpha, amax, n);
    pexp_k<<<(BEC * HH) / 256, 256, 0, stream>>>(alpha, amax, den, dl, em, n);
    hipMemsetAsync(agg, 0, (size_t)M * HD * 4, stream);
    agg_k<<<((unsigned)BEC * HD) / 256, 256, 0, stream>>>(vbuf, ebuf, alpha,
                                                          sl, dl, agg, n);
    gate_k<<<M / 4, 128, 0, stream>>>(agg, skip, den, cv[blk].Wb, q, M);

    gemm(q, tW[blk], tb[blk], xnext, M, DD, HD, 1);   // transf + relu
    hipMemsetAsync(st, 0, 2 * DD * 4, stream);
    bn_stats_k<<<128, 128, 0, stream>>>(xnext, st, M);
    bn_apply_k<<<((unsigned)M * DD) / 256, 256, 0, stream>>>(xnext, st,
                                                             bng[blk], bnb[blk], M);
    if (blk == 0) {
      xin = xnext; K = DD;
    } else {
      int layer = blk - 1;
      int kk = n / 2;
      float* xp = (xnext == xA) ? xB : xA;
      wnorm_k<<<1, 128, 0, stream>>>(poolw[layer], wn);
      score_k<<<M / 4, 128, 0, stream>>>(xnext, poolw[layer], wn, sc, M);
      topk_k<<<NB, 128, 0, stream>>>(sc, idxk, vals, inv, n, kk);
      pool_gather_k<<<NB * kk, 128, 0, stream>>>(xnext, idxk, vals, xp, n, kk);
      remap_k<<<BEC / 256, 256, 0, stream>>>(sl, dl, em, inv, n);
      reps_k<<<NB, 128, 0, stream>>>(xp, hrep, kk);
      n = kk; xin = xp; K = DD;
    }
  }

  // final MLP: 384 -> 256 -> 128 -> 1
  gemm(hrep, l1W, l1b, m1, NB, 256, 3 * DD, 1);
  gemm(m1,   l2W, l2b, m2, NB, DD, 256, 1);
  rowdot_k<<<NB / 4, 128, 0, stream>>>(m2, l3W, l3b, (float*)d_out, NB, DD);
}